// DeformableConv2d_66649302499761
// MI455X (gfx1250) — compile-verified
//
#include <hip/hip_runtime.h>

#define BATCH 4
#define CIN   256
#define COUT  256
#define HH    64
#define WW    64
#define HW    (HH*WW)          // 4096
#define KTAPS 9                // 3x3
#define OFFC  (2*KTAPS)        // 18 offset channels
#define KTOT  (CIN*KTAPS)      // 2304 reduction length
#define CCHUNK 8               // input channels per K-chunk
#define KCHUNK (CCHUNK*KTAPS)  // 72
#define APAD   73              // LDS row stride (pad for bank conflicts)

typedef float v2f __attribute__((ext_vector_type(2)));
typedef float v8f __attribute__((ext_vector_type(8)));

// ---------------------------------------------------------------------------
// Kernel 1: standard 3x3 conv producing the 18 offset channels (small: 1.4 GFLOP)
// One thread per output element (4*18*64*64 = 294912).
// ---------------------------------------------------------------------------
__global__ __launch_bounds__(256)
void offset_conv_kernel(const float* __restrict__ x,
                        const float* __restrict__ ow,
                        const float* __restrict__ obias,
                        float* __restrict__ off)
{
    int idx = blockIdx.x * 256 + threadIdx.x;       // grid sized exactly
    int b  = idx / (OFFC*HW);
    int r  = idx - b*(OFFC*HW);
    int oc = r / HW;
    int p  = r - oc*HW;
    int h  = p / WW;
    int w  = p - h*WW;

    float acc = obias[oc];
    const float* xb = x  + (size_t)b*CIN*HW;
    const float* wr = ow + (size_t)oc*CIN*KTAPS;

    for (int c = 0; c < CIN; ++c) {
        const float* xc = xb + c*HW;
        const float* wc = wr + c*KTAPS;
#pragma unroll
        for (int ky = 0; ky < 3; ++ky) {
            int yy = h + ky - 1;
            if (yy < 0 || yy >= HH) continue;
#pragma unroll
            for (int kx = 0; kx < 3; ++kx) {
                int xx = w + kx - 1;
                if (xx < 0 || xx >= WW) continue;
                acc = fmaf(xc[yy*WW + xx], wc[ky*3 + kx], acc);
            }
        }
    }
    off[idx] = acc;
}

// ---------------------------------------------------------------------------
// Kernel 2: deformable conv as implicit GEMM with V_WMMA_F32_16X16X4_F32.
//   M = 16 pixels per workgroup (one row segment, one image)
//   N = 128 output channels per workgroup (8 waves x 16)
//   K = 2304 looped in 72-wide chunks (8 cin x 9 taps)
// Bilinear samples are materialized once per chunk in LDS and shared by all
// 8 waves (8x gather reuse).
// ---------------------------------------------------------------------------
__global__ __launch_bounds__(256)
void deform_wmma_kernel(const float* __restrict__ x,
                        const float* __restrict__ weight,
                        const float* __restrict__ off,
                        float* __restrict__ out)
{
    __shared__ int   s_idx[16*KTAPS*4];   // clipped gather indices per (pixel, tap, corner)
    __shared__ float s_wgt[16*KTAPS*4];   // bilinear weight * validity mask
    __shared__ float s_A[16*APAD];        // A-chunk: 16 pixels x 72 K (padded)

    const int lane = threadIdx.x & 31;
    const int wave = threadIdx.x >> 5;
    const int row  = lane & 15;           // M-row (A) / N-col (B) within tile
    const int hi   = lane >> 4;           // 0: K=0,1  1: K=2,3 halves

    const int mBase = blockIdx.x * 16;    // flat pixel index (b*4096 + h*64 + w0)
    const int b     = mBase / HW;
    const int pix0  = mBase - b*HW;
    const int h     = pix0 / WW;
    const int w0    = pix0 - h*WW;        // 16 consecutive pixels in one row
    const int nBase = blockIdx.y * 128 + wave * 16;

    // ---- per-(pixel, tap) bilinear geometry (threads 0..143) ----
    if (threadIdx.x < 16*KTAPS) {
        int m  = threadIdx.x / KTAPS;
        int kk = threadIdx.x - m*KTAPS;
        int wcol = w0 + m;
        const float* ob = off + (size_t)b*OFFC*HW + h*WW + wcol;
        float dy = ob[(2*kk + 0)*HW];
        float dx = ob[(2*kk + 1)*HW];
        float py = (float)(h    - 1 + kk/3) + dy;
        float px = (float)(wcol - 1 + kk%3) + dx;
        float y0f = floorf(py), x0f = floorf(px);
        int   y0  = (int)y0f,  x0  = (int)x0f;
        float wy1 = py - y0f, wy0 = 1.0f - wy1;
        float wx1 = px - x0f, wx0 = 1.0f - wx1;

        int   ys[2] = { y0, y0 + 1 };
        int   xs[2] = { x0, x0 + 1 };
        float wys[2] = { wy0, wy1 };
        float wxs[2] = { wx0, wx1 };
        int base = (m*KTAPS + kk)*4;
#pragma unroll
        for (int j = 0; j < 4; ++j) {
            int yy = ys[j >> 1], xx = xs[j & 1];
            bool valid = (yy >= 0) & (yy < HH) & (xx >= 0) & (xx < WW);
            int yc = min(max(yy, 0), HH-1);
            int xc = min(max(xx, 0), WW-1);
            s_idx[base + j] = yc*WW + xc;
            s_wgt[base + j] = valid ? wys[j >> 1]*wxs[j & 1] : 0.0f;
        }
    }
    __syncthreads();

    v8f acc = {0.f,0.f,0.f,0.f,0.f,0.f,0.f,0.f};
    const float* xb = x + (size_t)b*CIN*HW;
    const float* wn = weight + (size_t)(nBase + row)*KTOT;   // this lane's N column

    for (int cc = 0; cc < CIN; cc += CCHUNK) {
        // ---- cooperative fill of the 16x72 A-chunk (bilinear samples) ----
        for (int e = threadIdx.x; e < 16*KCHUNK; e += 256) {
            int m  = e / KCHUNK;
            int kl = e - m*KCHUNK;
            int cl = kl / KTAPS;
            int kk = kl - cl*KTAPS;
            const float* xc = xb + (size_t)(cc + cl)*HW;
            int g = (m*KTAPS + kk)*4;
            float v = s_wgt[g+0]*xc[s_idx[g+0]]
                    + s_wgt[g+1]*xc[s_idx[g+1]]
                    + s_wgt[g+2]*xc[s_idx[g+2]]
                    + s_wgt[g+3]*xc[s_idx[g+3]];
            s_A[m*APAD + kl] = v;
        }
        __syncthreads();

        // ---- 18 x V_WMMA_F32_16X16X4_F32 over this 72-wide K chunk ----
        const float* wk = wn + cc*KTAPS;
#pragma unroll
        for (int kb = 0; kb < KCHUNK; kb += 4) {
            v2f a, bf;
            // A fragment: lane(row) holds A[M=row][K=kb+2*hi .. +1]
            a.x = s_A[row*APAD + kb + 2*hi + 0];
            a.y = s_A[row*APAD + kb + 2*hi + 1];
            // B fragment: lane(row=N col) holds B[K=kb+2*hi .. +1][N]
            bf.x = wk[kb + 2*hi + 0];
            bf.y = wk[kb + 2*hi + 1];
            acc = __builtin_amdgcn_wmma_f32_16x16x4_f32(
                    /*neg_a=*/false, a, /*neg_b=*/false, bf,
                    /*c_mod=*/(short)0, acc, /*reuse_a=*/false, /*reuse_b=*/false);
        }
        __syncthreads();
    }

    // ---- store: VGPR r -> M = r (lanes 0-15) / 8+r (lanes 16-31), N = lane&15 ----
    float* obp = out + ((size_t)b*COUT + nBase + row)*HW + h*WW + w0;
#pragma unroll
    for (int r = 0; r < 8; ++r) {
        int mloc = hi ? (8 + r) : r;
        obp[mloc] = acc[r];
    }
}

// ---------------------------------------------------------------------------
extern "C" void kernel_launch(void* const* d_in, const int* in_sizes, int n_in,
                              void* d_out, int out_size, void* d_ws, size_t ws_size,
                              hipStream_t stream)
{
    const float* x      = (const float*)d_in[0];
    const float* ow     = (const float*)d_in[1];
    const float* obias  = (const float*)d_in[2];
    const float* weight = (const float*)d_in[3];
    float* off = (float*)d_ws;              // 4*18*64*64 floats = 1.13 MB
    float* out = (float*)d_out;

    // 1) offset conv: 294912 elements / 256 = 1152 blocks
    offset_conv_kernel<<<dim3((BATCH*OFFC*HW)/256), 256, 0, stream>>>(x, ow, obias, off);

    // 2) deformable conv GEMM: (16384/16) M-tiles x 2 N-halves
    deform_wmma_kernel<<<dim3((BATCH*HW)/16, COUT/128), 256, 0, stream>>>(x, weight, off, out);
}